// MyGCN_v3_5102421148072
// MI455X (gfx1250) — compile-verified
//
#include <hip/hip_runtime.h>
#include <hip/hip_bf16.h>

typedef __attribute__((ext_vector_type(2))) float v2f;
typedef __attribute__((ext_vector_type(8))) float v8f;

#define NPAD 16    // all hidden widths padded to 16 columns
#define KMAX 512   // max K across layers (layer 1)

// One wave per 16-row output tile: C[16 x 16] = A[16 x K4] * W[K x M] (padded),
// f32 end-to-end via V_WMMA_F32_16X16X4_F32 (exact precision match).
// W is zero-padded into LDS (K4 x 16) once per block and shared by 8 waves:
// inner loop is 1 global_load_b64 (A) + 2 conflict-free ds_load_b32 (B) + WMMA.
__global__ __launch_bounds__(256) void gcn_gemm_wmma(
    const float* __restrict__ A, const float* __restrict__ B,
    float* __restrict__ C, int N, int K, int M, int K4, int lda) {
  __shared__ float sB[KMAX * NPAD];

  // Cooperative zero-padded fill: sB[k][f] = (k<K && f<M) ? B[k*M+f] : 0
  for (int t = threadIdx.x; t < K4 * NPAD; t += 256) {
    int k = t >> 4;
    int f = t & (NPAD - 1);
    sB[t] = (k < K && f < M) ? B[k * M + f] : 0.0f;
  }
  __syncthreads();

  int wave = (blockIdx.x * blockDim.x + threadIdx.x) >> 5;
  int lane = threadIdx.x & 31;
  int row0 = wave * 16;
  if (row0 >= N) return;  // after the barrier: every wave participated above

  int half = lane >> 4;   // 0: K pair {k0,k0+1}; 1: K pair {k0+2,k0+3}
  int l    = lane & 15;   // A: row-in-tile ; B/C/D: column

  int m = row0 + l;
  if (m >= N) m = N - 1;  // clamp (partial tiles only; N%16==0 in practice)

  const float* arow = A + (size_t)m * lda + 2 * half;    // {k0+2h, k0+2h+1}
  const float* bcol = sB + 32 * half + l;                // sB[2h][l]

  v8f acc = {};
#pragma unroll 4
  for (int k0 = 0; k0 < K4; k0 += 4) {
    v2f a = *(const v2f*)arow;      // one global_load_b64
    v2f bf;
    bf.x = bcol[0];                 // sB[k0+2h  ][l]   (ds_load_b32)
    bf.y = bcol[NPAD];              // sB[k0+2h+1][l]   (ds_load_b32)
    // 8 args: (neg_a, A, neg_b, B, c_mod, C, reuse_a, reuse_b)
    acc = __builtin_amdgcn_wmma_f32_16x16x4_f32(
        false, a, false, bf, (short)0, acc, false, false);
    arow += 4;
    bcol += 4 * NPAD;
  }

  // D layout: lanes 0-15 hold rows 0..7 in acc[0..7]; lanes 16-31 rows 8..15.
  if (row0 + 16 <= N) {             // wave-uniform fast path: unguarded stores
#pragma unroll
    for (int r = 0; r < 8; ++r)
      C[(size_t)(row0 + 8 * half + r) * NPAD + l] = acc[r];
  } else {
#pragma unroll
    for (int r = 0; r < 8; ++r) {
      int mm = row0 + 8 * half + r;
      if (mm < N) C[(size_t)mm * NPAD + l] = acc[r];
    }
  }
}

// agg[dst, f] += w[e] * support[src, f]; one thread per (edge, padded feature).
// 16 consecutive lanes share one edge -> coalesced gather & atomic bursts.
__global__ __launch_bounds__(256) void gcn_scatter(
    const int* __restrict__ src, const int* __restrict__ dst,
    const float* __restrict__ w, const float* __restrict__ support,
    float* __restrict__ agg, int nE, int M) {
  long long tid = (long long)blockIdx.x * blockDim.x + threadIdx.x;
  long long total = (long long)nE * NPAD;
  if (tid >= total) return;
  int e = (int)(tid >> 4);
  int f = (int)(tid & (NPAD - 1));
  if (f < M) {
    int s = src[e];
    int d = dst[e];
    float v = w[e] * support[(size_t)s * NPAD + f];
    atomicAdd(&agg[(size_t)d * NPAD + f], v);
  }
}

// out[n, f] = agg[n, f] + b[f] for f < M, else 0 (keeps pad zeroed for next layer).
__global__ __launch_bounds__(256) void gcn_bias(
    const float* __restrict__ agg, const float* __restrict__ b,
    float* __restrict__ out, long long total, int M) {
  long long tid = (long long)blockIdx.x * blockDim.x + threadIdx.x;
  if (tid >= total) return;
  int f = (int)(tid & (NPAD - 1));
  out[tid] = (f < M) ? (agg[tid] + b[f]) : 0.0f;
}

extern "C" void kernel_launch(void* const* d_in, const int* in_sizes, int n_in,
                              void* d_out, int out_size, void* d_ws, size_t ws_size,
                              hipStream_t stream) {
  const float* x          = (const float*)d_in[0];
  const int*   edge_index = (const int*)d_in[1];
  const float* ew         = (const float*)d_in[2];
  const float* W[6];
  const float* b[6];
  for (int i = 0; i < 6; ++i) {
    W[i] = (const float*)d_in[3 + 2 * i];
    b[i] = (const float*)d_in[4 + 2 * i];
  }

  const int NFEAT = 512;
  int N = in_sizes[0] / NFEAT;   // 100000
  int E = in_sizes[1] / 2;       // 3200000
  const int* src = edge_index;       // edge_index[0, :]
  const int* dst = edge_index + E;   // edge_index[1, :]

  // Workspace: three padded N x 16 f32 buffers (~19.2 MB total).
  float* support = (float*)d_ws;
  float* agg     = support + (size_t)N * NPAD;
  float* h       = agg     + (size_t)N * NPAD;

  const int dims[7] = {512, 12, 10, 8, 6, 4, 16};

  const float* curA = x;
  int lda = NFEAT;
  long long nodes_total = (long long)N * NPAD;

  for (int i = 0; i < 6; ++i) {
    int K  = dims[i];
    int M  = dims[i + 1];
    int K4 = (K + 3) & ~3;

    // 1) support = curA @ W[i]  (WMMA, one wave per 16-row tile, 8 waves/block)
    int tiles  = (N + 15) / 16;
    int blocks = (tiles + 7) / 8;
    gcn_gemm_wmma<<<blocks, 256, 0, stream>>>(curA, W[i], support, N, K, M, K4, lda);

    // 2) agg = 0
    hipMemsetAsync(agg, 0, (size_t)N * NPAD * sizeof(float), stream);

    // 3) agg[dst] += w * support[src]
    long long etotal  = (long long)E * NPAD;
    int       eblocks = (int)((etotal + 255) / 256);
    gcn_scatter<<<eblocks, 256, 0, stream>>>(src, dst, ew, support, agg, E, M);

    // 4) h_next = agg + b (last layer writes d_out; M==16 there, no padding)
    float* outp = (i == 5) ? (float*)d_out : h;
    int nblocks = (int)((nodes_total + 255) / 256);
    gcn_bias<<<nblocks, 256, 0, stream>>>(agg, b[i], outp, nodes_total, M);

    curA = h;
    lda  = NPAD;
  }
}